// GCN_26817775797032
// MI455X (gfx1250) — compile-verified
//
#include <hip/hip_runtime.h>

// ---------------------------------------------------------------------------
// GCN (3-layer) for MI455X / gfx1250.
//   norm:      deg (with self loops) -> dis = rsqrt(deg)
//   per layer: H = X @ W  (V_WMMA_F32_16X16X4_F32, exact fp32)
//              agg[dst] += H[src] * dis[src]*dis[dst]   (fp32 atomics, L2-resident)
//              out = relu(agg + H*dis^2 + b)            (self-loop fused here)
// ---------------------------------------------------------------------------

typedef __attribute__((ext_vector_type(2))) float v2f;
typedef __attribute__((ext_vector_type(8))) float v8f;

#define GCN_N_NODES 50000

// ---------------- normalization ----------------
__global__ void deg_init_kernel(unsigned* __restrict__ deg, int n) {
  int i = blockIdx.x * blockDim.x + threadIdx.x;
  if (i < n) deg[i] = 1u;                       // self-loop contributes 1
}

__global__ void deg_count_kernel(const int* __restrict__ dst,
                                 unsigned* __restrict__ deg, int nE) {
  int i = blockIdx.x * blockDim.x + threadIdx.x;
  if (i < nE) atomicAdd(&deg[dst[i]], 1u);
}

__global__ void dis_kernel(const unsigned* __restrict__ deg,
                           float* __restrict__ dis, int n) {
  int i = blockIdx.x * blockDim.x + threadIdx.x;
  if (i < n) dis[i] = rsqrtf((float)deg[i]);    // deg >= 1 always
}

__global__ void zero_kernel(float* __restrict__ p, long n) {
  long i = (long)blockIdx.x * blockDim.x + threadIdx.x;
  if (i < n) p[i] = 0.0f;
}

// ---------------- fp32 WMMA GEMM: H[M,NCOL] = A[M,K] @ W[K,NCOL] ----------
// One wave per 16x16 output tile. grid = (M/16, NCOL/16), block = 32 (wave32).
// ISA 7.12.2 f32 layouts:
//   A 16x4  : VGPR0 = K=0 (lanes 0-15) / K=2 (lanes 16-31); VGPR1 = K=1 / K=3
//   B 4x16  : VGPR0 = row K=0 (lanes 0-15) / K=2 (lanes 16-31); VGPR1 = K=1/K=3
//   C/D 16x16: VGPR r = row r (lanes 0-15) / row r+8 (lanes 16-31)
template<int K, int NCOL>
__global__ void gemm_wmma_kernel(const float* __restrict__ A,
                                 const float* __restrict__ W,
                                 float* __restrict__ H) {
  const int lane = threadIdx.x & 31;
  const int half = lane >> 4;      // 0 or 1
  const int l    = lane & 15;
  const int m0   = blockIdx.x << 4;
  const int n0   = blockIdx.y << 4;

  v8f acc = {};
  const float* arow = A + (long)(m0 + l) * K + 2 * half;   // contiguous pair -> b64 load
  const float* bcol = W + (long)(2 * half) * NCOL + n0 + l;

  #pragma unroll 4
  for (int k0 = 0; k0 < K; k0 += 4) {
    v2f a; a.x = arow[0]; a.y = arow[1];
    v2f b; b.x = bcol[0]; b.y = bcol[NCOL];
    acc = __builtin_amdgcn_wmma_f32_16x16x4_f32(
        /*neg_a=*/false, a, /*neg_b=*/false, b,
        /*c_mod=*/(short)0, acc, /*reuse_a=*/false, /*reuse_b=*/false);
    arow += 4;
    bcol += (long)4 * NCOL;
  }

  float* hp = H + (long)(m0 + 8 * half) * NCOL + n0 + l;
  #pragma unroll
  for (int r = 0; r < 8; ++r) hp[(long)r * NCOL] = acc[r];
}

// ---------------- edge scatter: agg[dst] += H[src] * dis[src]*dis[dst] -----
// One wave per edge; lanes stride the F features (coalesced loads, coalesced
// GLOBAL_ATOMIC_ADD_F32). H rows are L2-resident (25.6MB << 192MB L2).
template<int F>
__global__ void scatter_kernel(const float* __restrict__ h,
                               const int* __restrict__ src,
                               const int* __restrict__ dst,
                               const float* __restrict__ dis,
                               float* __restrict__ agg, int nE) {
  int wv   = (int)(((long)blockIdx.x * blockDim.x + threadIdx.x) >> 5);
  int lane = threadIdx.x & 31;
  if (wv >= nE) return;
  int s = src[wv];
  int d = dst[wv];
  float nrm = dis[s] * dis[d];
  const float* hs = h + (long)s * F;
  float* od = agg + (long)d * F;
  #pragma unroll
  for (int f = lane; f < F; f += 32)
    atomicAdd(od + f, hs[f] * nrm);
}

// ---------------- finalize: out = (relu?)(agg + H*dis^2 + b) ---------------
template<int F, bool RELU>
__global__ void finalize_kernel(const float* __restrict__ agg,
                                const float* __restrict__ h,
                                const float* __restrict__ dis,
                                const float* __restrict__ bias,
                                float* __restrict__ out, int nN) {
  long t = (long)blockIdx.x * blockDim.x + threadIdx.x;
  int node = (int)(t >> __builtin_ctz(F));     // F is a power of two
  int f    = (int)(t & (F - 1));
  if (node >= nN) return;
  float di = dis[node];
  float v = agg[t] + h[t] * (di * di) + bias[f];
  out[t] = RELU ? fmaxf(v, 0.0f) : v;
}

// ---------------------------------------------------------------------------
extern "C" void kernel_launch(void* const* d_in, const int* in_sizes, int n_in,
                              void* d_out, int out_size, void* d_ws, size_t ws_size,
                              hipStream_t stream) {
  const float* x  = (const float*)d_in[0];
  const int*   ei = (const int*)d_in[1];   // int32 on device (jax x64 off)
  const float* W1 = (const float*)d_in[2];
  const float* b1 = (const float*)d_in[3];
  const float* W2 = (const float*)d_in[4];
  const float* b2 = (const float*)d_in[5];
  const float* W3 = (const float*)d_in[6];
  const float* b3 = (const float*)d_in[7];
  float* out = (float*)d_out;

  const int nN = GCN_N_NODES;
  const int nE = in_sizes[1] / 2;          // edge_index is [2, E] row-major
  const int* src = ei;
  const int* dst = ei + nE;

  // workspace carve-up
  char* ws = (char*)d_ws;
  unsigned* deg = (unsigned*)ws;  ws += (size_t)nN * 4;
  float* dis    = (float*)ws;     ws += (size_t)nN * 4;
  float* bufA   = (float*)ws;     ws += (size_t)nN * 128 * 4;   // H / act (in-place)
  float* bufB   = (float*)ws;     ws += (size_t)nN * 128 * 4;   // agg accumulator
  float* bufC   = (float*)ws;     ws += (size_t)nN * 128 * 4;   // second H / act

  auto cdiv = [](long a, long b) { return (unsigned)((a + b - 1) / b); };
  const long n128 = (long)nN * 128;
  const long n16  = (long)nN * 16;
  const long eThreads = (long)nE * 32;

  // ---- symmetric normalization ----
  deg_init_kernel <<<cdiv(nN, 256), 256, 0, stream>>>(deg, nN);
  deg_count_kernel<<<cdiv(nE, 256), 256, 0, stream>>>(dst, deg, nE);
  dis_kernel      <<<cdiv(nN, 256), 256, 0, stream>>>(deg, dis, nN);

  // ---- layer 1: 256 -> 128, ReLU ----
  gemm_wmma_kernel<256, 128><<<dim3(nN / 16, 128 / 16), 32, 0, stream>>>(x, W1, bufA);
  zero_kernel     <<<cdiv(n128, 256), 256, 0, stream>>>(bufB, n128);
  scatter_kernel<128><<<cdiv(eThreads, 256), 256, 0, stream>>>(bufA, src, dst, dis, bufB, nE);
  finalize_kernel<128, true><<<cdiv(n128, 256), 256, 0, stream>>>(bufB, bufA, dis, b1, bufA, nN);

  // ---- layer 2: 128 -> 128, ReLU ----
  gemm_wmma_kernel<128, 128><<<dim3(nN / 16, 128 / 16), 32, 0, stream>>>(bufA, W2, bufC);
  zero_kernel     <<<cdiv(n128, 256), 256, 0, stream>>>(bufB, n128);
  scatter_kernel<128><<<cdiv(eThreads, 256), 256, 0, stream>>>(bufC, src, dst, dis, bufB, nE);
  finalize_kernel<128, true><<<cdiv(n128, 256), 256, 0, stream>>>(bufB, bufC, dis, b2, bufC, nN);

  // ---- layer 3: 128 -> 16, no ReLU, writes d_out ----
  gemm_wmma_kernel<128, 16><<<dim3(nN / 16, 1), 32, 0, stream>>>(bufC, W3, bufA);
  zero_kernel     <<<cdiv(n16, 256), 256, 0, stream>>>(bufB, n16);
  scatter_kernel<16><<<cdiv(eThreads, 256), 256, 0, stream>>>(bufA, src, dst, dis, bufB, nE);
  finalize_kernel<16, false><<<cdiv(n16, 256), 256, 0, stream>>>(bufB, bufA, dis, b3, out, nN);
}